// EGraphSage_49280454754450
// MI455X (gfx1250) — compile-verified
//
#include <hip/hip_runtime.h>

typedef __attribute__((ext_vector_type(2))) float v2f;
typedef __attribute__((ext_vector_type(4))) float v4f;
typedef __attribute__((ext_vector_type(8))) float v8f;

#define EMBED 128
#define FEAT  64
#define EE_W  320            // 2*EMBED + FEAT
#define CLS   2
#define LDS_STRIDE 328       // padded row stride (floats) to avoid bank-0 pileup
#define WAVES 4
#define TPB   (WAVES * 32)
#define EDGES_PER_WAVE 16
#define EDGES_PER_BLOCK (WAVES * EDGES_PER_WAVE)

// Streaming (write-once) outputs: non-temporal stores keep the 51 MB
// node-embed gather table resident in the 192 MB L2.
__device__ __forceinline__ void store_nt4(float* p, v4f v) {
    __builtin_nontemporal_store(v, (v4f*)p);
}

__global__ __launch_bounds__(TPB)
void egsage_gather_score_kernel(const float* __restrict__ node,
                                const float* __restrict__ efeat,
                                const float* __restrict__ weight,
                                const int*   __restrict__ src,
                                const int*   __restrict__ dst,
                                float*       __restrict__ scores,
                                float*       __restrict__ ee,
                                int E)
{
    __shared__ float sW[16 * LDS_STRIDE];               // zero-padded W (16x320)
    __shared__ float sEE[WAVES * 16 * LDS_STRIDE];      // per-wave 16x320 ee tiles

    const int tid = threadIdx.x;

    // ---- stage weight (2 x 320) into a zero-padded 16 x 328 tile ----
    for (int idx = tid; idx < 16 * LDS_STRIDE; idx += TPB) {
        int r = idx / LDS_STRIDE;
        int c = idx - r * LDS_STRIDE;
        float v = 0.0f;
        if (r < CLS && c < EE_W) v = weight[r * EE_W + c];
        sW[idx] = v;
    }
    __syncthreads();

    const int lane = tid & 31;
    const int wave = tid >> 5;
    const int e0   = blockIdx.x * EDGES_PER_BLOCK + wave * EDGES_PER_WAVE;
    float* tile = &sEE[wave * 16 * LDS_STRIDE];

    // ---- Phase A: gather src embeddings -> cols [0,128), stream to ee out ----
    for (int i = 0; i < 16; ++i) {
        int e = e0 + i;
        bool ok = (e < E);
        v4f v = {0.f, 0.f, 0.f, 0.f};
        if (ok) {
            int s = src[e];
            v = *(const v4f*)&node[(long)s * EMBED + lane * 4];
        }
        *(v4f*)&tile[i * LDS_STRIDE + lane * 4] = v;
        if (ok) store_nt4(&ee[(long)e * EE_W + lane * 4], v);
    }
    // ---- Phase B: gather dst embeddings -> cols [128,256) ----
    for (int i = 0; i < 16; ++i) {
        int e = e0 + i;
        bool ok = (e < E);
        v4f v = {0.f, 0.f, 0.f, 0.f};
        if (ok) {
            int d = dst[e];
            v = *(const v4f*)&node[(long)d * EMBED + lane * 4];
        }
        *(v4f*)&tile[i * LDS_STRIDE + EMBED + lane * 4] = v;
        if (ok) store_nt4(&ee[(long)e * EE_W + EMBED + lane * 4], v);
    }
    // ---- Phase C: edge features -> cols [256,320); 2 rows per iteration ----
    for (int i = 0; i < 8; ++i) {
        int r  = i * 2 + (lane >> 4);
        int c4 = lane & 15;
        int e  = e0 + r;
        bool ok = (e < E);
        v4f v = {0.f, 0.f, 0.f, 0.f};
        if (ok) v = *(const v4f*)&efeat[(long)e * FEAT + c4 * 4];
        *(v4f*)&tile[r * LDS_STRIDE + 2 * EMBED + c4 * 4] = v;
        if (ok) store_nt4(&ee[(long)e * EE_W + 2 * EMBED + c4 * 4], v);
    }

    // ---- WMMA: D(16x16) += A(16x4,f32) x B(4x16,f32), 80 chained K-steps ----
    // A frag: lane L -> {ee[L&15][k0 + 2*(L>>4)], +1}   (ds_load_b64)
    // B frag: lane L -> {W [L&15][k0 + 2*(L>>4)], +1}   (zero rows for cls>=2)
    const int m  = lane & 15;
    const int kk = (lane >> 4) * 2;
    v8f acc = {0.f, 0.f, 0.f, 0.f, 0.f, 0.f, 0.f, 0.f};
    for (int k0 = 0; k0 < EE_W; k0 += 4) {
        v2f a = *(const v2f*)&tile[m * LDS_STRIDE + k0 + kk];
        v2f b = *(const v2f*)&sW  [m * LDS_STRIDE + k0 + kk];
        acc = __builtin_amdgcn_wmma_f32_16x16x4_f32(
            /*neg_a=*/false, a, /*neg_b=*/false, b,
            /*c_mod=*/(short)0, acc, /*reuse_a=*/false, /*reuse_b=*/false);
    }

    // ---- D layout: VGPR v, lanes 0-15: M=v, N=lane; lanes 16-31: M=v+8, N=lane-16
    const int n     = lane & 15;
    const int mbase = (lane >> 4) * 8;
    if (n < CLS) {
        #pragma unroll
        for (int v = 0; v < 8; ++v) {
            int e = e0 + mbase + v;
            if (e < E) __builtin_nontemporal_store(acc[v], &scores[(long)e * CLS + n]);
        }
    }
}

extern "C" void kernel_launch(void* const* d_in, const int* in_sizes, int n_in,
                              void* d_out, int out_size, void* d_ws, size_t ws_size,
                              hipStream_t stream) {
    const float* node   = (const float*)d_in[0];   // [N, 128]
    const float* efeat  = (const float*)d_in[1];   // [E, 64]
    const float* weight = (const float*)d_in[2];   // [2, 320]
    const int*   src    = (const int*)d_in[3];     // [E]
    const int*   dst    = (const int*)d_in[4];     // [E]
    const int E = in_sizes[3];

    float* scores = (float*)d_out;                 // [E, 2]
    float* ee     = (float*)d_out + (long)E * CLS; // [E, 320]

    const int nblocks = (E + EDGES_PER_BLOCK - 1) / EDGES_PER_BLOCK;
    egsage_gather_score_kernel<<<nblocks, TPB, 0, stream>>>(
        node, efeat, weight, src, dst, scores, ee, E);
}